// BSEC_RNN_68839735820382
// MI455X (gfx1250) — compile-verified
//
#include <hip/hip_runtime.h>
#include <cmath>
#include <stdint.h>

// Problem constants (match reference)
#define NB 4096   // batch
#define TT 2048   // timesteps
#define BB 64     // batch rows per block (1 thread = 1 chain)
#define TC 32     // timesteps per LDS chunk
#define NC (TT / TC)        // 64 chunks
#define RS 36     // LDS row stride in floats (padded, 16B-aligned, 2-way banking)

// ---- CDNA5 feature detection -------------------------------------------
#if __has_builtin(__builtin_amdgcn_global_load_async_to_lds_b128)
#define HAVE_ASYNC_LDS 1
#else
#define HAVE_ASYNC_LDS 0
#endif

#if HAVE_ASYNC_LDS
// Exact parameter types per hipcc diagnostic:
//   arg0: int __attribute__((vector_size(16))) addrspace(1)*   (global, non-const)
//   arg1: (expected) same vector type in addrspace(3)          (LDS)
typedef int v4i_t __attribute__((__vector_size__(16)));
typedef __attribute__((address_space(1))) v4i_t* as1_v4i;
typedef __attribute__((address_space(3))) v4i_t* as3_v4i;
#if __has_builtin(__builtin_amdgcn_s_wait_asynccnt)
#define WAIT_ASYNC(n) __builtin_amdgcn_s_wait_asynccnt(n)
#else
#define WAIT_ASYNC(n) asm volatile("s_wait_asynccnt %0" ::"i"(n) : "memory")
#endif
#else
#define WAIT_ASYNC(n)
#endif

__device__ __forceinline__ float fast_tanh(float v) {
#if __has_builtin(__builtin_amdgcn_tanhf)
    return __builtin_amdgcn_tanhf(v);          // gfx1250 v_tanh_f32
#elif __has_builtin(__builtin_amdgcn_tanh_f32)
    return __builtin_amdgcn_tanh_f32(v);
#else
    return tanhf(v);
#endif
}

__global__ __launch_bounds__(BB) void BSEC_RNN_68839735820382_kernel(
    const float* __restrict__ x,       // [B, T, 1]
    const float* __restrict__ W_ih,    // [2, 1]
    const float* __restrict__ W_hh,    // [2, 2]
    const float* __restrict__ b_ih,    // [2]
    const float* __restrict__ b_hh,    // [2]
    const float* __restrict__ fc_w,    // [1, 3]  (coeffs for x, h0, h1)
    const float* __restrict__ fc_b,    // [1]
    float* __restrict__ out)           // [B, T, 1]
{
    __shared__ __align__(16) float inbuf[2][BB * RS];
    __shared__ __align__(16) float outbuf[2][BB * RS];

    const int tid   = threadIdx.x;
    const int bBase = blockIdx.x * BB;

    // Uniform weights -> scalar registers
    const float wih0 = W_ih[0], wih1 = W_ih[1];
    const float w00 = W_hh[0], w01 = W_hh[1], w10 = W_hh[2], w11 = W_hh[3];
    const float bias0 = b_ih[0] + b_hh[0];
    const float bias1 = b_ih[1] + b_hh[1];
    const float fc0 = fc_w[0], fc1 = fc_w[1], fc2 = fc_w[2], fcb = fc_b[0];

    // Coalesced tile stage: tile is [BB rows x TC floats]; BB*TC/4 float4
    // slots, 8 per thread; consecutive threads touch consecutive 16B chunks.
    auto stage = [&](int chunk, int buf) {
        const int t0 = chunk * TC;
#pragma unroll
        for (int s = 0; s < 8; ++s) {
            const int idx = s * BB + tid;
            const int row = idx >> 3;
            const int c4  = (idx & 7) * 4;
            const float* g = x + (size_t)(bBase + row) * TT + t0 + c4;
            float* l = &inbuf[buf][row * RS + c4];
#if HAVE_ASYNC_LDS
            __builtin_amdgcn_global_load_async_to_lds_b128(
                (as1_v4i)(uintptr_t)g,   // global src (int cast: sheds const cleanly)
                (as3_v4i)l,              // LDS dst (explicit generic->local cast)
                0, 0);
#else
            *(float4*)l = *(const float4*)g;
#endif
        }
    };

    float h0 = 0.0f, h1 = 0.0f;

    stage(0, 0);  // prefetch chunk 0

    for (int k = 0; k < NC; ++k) {
        const int cur = k & 1;
        if (k + 1 < NC) stage(k + 1, cur ^ 1);  // prefetch ahead (8 async ops/wave)

        // Async loads complete in order: <=8 outstanding => chunk k's 8 done.
        if (k + 1 < NC) { WAIT_ASYNC(8); } else { WAIT_ASYNC(0); }
        __syncthreads();

        const float* xr  = &inbuf[cur][tid * RS];
        float*       orw = &outbuf[cur][tid * RS];
#pragma unroll
        for (int j = 0; j < 8; ++j) {
            const float4 xq = *(const float4*)&xr[4 * j];
            float4 oq;
            {   // 4 dependent recurrence steps
                const float xv[4] = {xq.x, xq.y, xq.z, xq.w};
                float ov[4];
#pragma unroll
                for (int q = 0; q < 4; ++q) {
                    const float p0 = fmaf(xv[q], wih0, bias0);   // off-chain
                    const float p1 = fmaf(xv[q], wih1, bias1);   // off-chain
                    const float n0 = fmaf(h0, w00, fmaf(h1, w01, p0));
                    const float n1 = fmaf(h0, w10, fmaf(h1, w11, p1));
                    h0 = fast_tanh(n0);
                    h1 = fast_tanh(n1);
                    ov[q] = fmaf(h0, fc1, fmaf(h1, fc2, fmaf(xv[q], fc0, fcb)));
                }
                oq = make_float4(ov[0], ov[1], ov[2], ov[3]);
            }
            *(float4*)&orw[4 * j] = oq;
        }
        __syncthreads();

        // Coalesced writeback of the staged output tile
        const int t0 = k * TC;
#pragma unroll
        for (int s = 0; s < 8; ++s) {
            const int idx = s * BB + tid;
            const int row = idx >> 3;
            const int c4  = (idx & 7) * 4;
            const float4 v = *(const float4*)&outbuf[cur][row * RS + c4];
            *(float4*)(out + (size_t)(bBase + row) * TT + t0 + c4) = v;
        }
        // next iteration's leading __syncthreads protects outbuf reuse
    }
}

extern "C" void kernel_launch(void* const* d_in, const int* in_sizes, int n_in,
                              void* d_out, int out_size, void* d_ws, size_t ws_size,
                              hipStream_t stream) {
    (void)in_sizes; (void)n_in; (void)d_ws; (void)ws_size; (void)out_size;
    const float* x    = (const float*)d_in[0];
    const float* W_ih = (const float*)d_in[1];
    const float* W_hh = (const float*)d_in[2];
    const float* b_ih = (const float*)d_in[3];
    const float* b_hh = (const float*)d_in[4];
    const float* fc_w = (const float*)d_in[5];
    const float* fc_b = (const float*)d_in[6];
    float* out = (float*)d_out;

    dim3 grid(NB / BB);   // 64 blocks
    dim3 block(BB);       // 64 threads, 2 waves
    BSEC_RNN_68839735820382_kernel<<<grid, block, 0, stream>>>(
        x, W_ih, W_hh, b_ih, b_hh, fc_w, fc_b, out);
}